// SleepEventClassificationLayer_52166672778116
// MI455X (gfx1250) — compile-verified
//
#include <hip/hip_runtime.h>
#include <stdint.h>
#include <stddef.h>

// ---------------------------------------------------------------------------
// SleepEventClassification for MI455X (gfx1250)
// Memory-bound streaming kernel: 175 MB in / 32 KB out -> HBM-limited (~7.5us).
// CDNA5 paths used:
//   * TENSOR_LOAD_TO_LDS (TDM DMA, 1 descriptor for half the batch block)
//   * GLOBAL_LOAD_ASYNC_TO_LDS_B64 (async mover for the other half)
//   * s_wait_tensorcnt / s_wait_asynccnt dependency counters
// ---------------------------------------------------------------------------

#define NFRAMES      42
#define NCLASS       521
#define BATCH_FLOATS (NFRAMES * NCLASS)   // 21882 floats = 87528 bytes (8B aligned per batch)
#define NTHREADS     256
#define NWAVES       (NTHREADS / 32)

#if defined(__AMDGCN__) && __has_builtin(__builtin_amdgcn_tensor_load_to_lds) && \
    __has_builtin(__builtin_amdgcn_s_wait_tensorcnt)
#define HAVE_TDM 1
#else
#define HAVE_TDM 0
#endif

#if HAVE_TDM
#define TDM_FLOATS 10942                  // first chunk via TDM (even -> 8B-aligned split)
#else
#define TDM_FLOATS 0
#endif
#define ASYNC_FLOATS (BATCH_FLOATS - TDM_FLOATS)
#define ASYNC_B64    (ASYNC_FLOATS / 2)

// 13 combos: class / lo / hi  (from BRUX_COMBOS)
__constant__ int kCls[13][3] = {
  {500, 41,  0}, {369,500,  0}, {412,431,470}, {127, 67,103},
  {399,403,412}, {410,411,398}, {412,410,398}, {435,438,449},
  {374,436,435}, {372,434,469}, {500,439, 50}, {399,403,410},
  {410,398,435}};
__constant__ float kLo[13][3] = {
  {0.10f,0.05f,0.00f},{0.10f,0.10f,0.00f},{0.05f,0.05f,0.05f},{0.10f,0.10f,0.10f},
  {0.10f,0.10f,0.10f},{0.10f,0.10f,0.10f},{0.10f,0.10f,0.10f},{0.10f,0.10f,0.10f},
  {0.10f,0.10f,0.10f},{0.10f,0.10f,0.10f},{0.10f,0.10f,0.10f},{0.10f,0.10f,0.10f},
  {0.10f,0.10f,0.10f}};
__constant__ float kHi[13][3] = {
  {0.8f,0.5f,1.0f},{0.5f,0.5f,1.0f},{0.4f,0.4f,0.4f},{0.8f,0.7f,0.6f},
  {0.9f,0.8f,0.7f},{0.9f,0.8f,0.7f},{0.8f,0.7f,0.6f},{0.9f,0.8f,0.7f},
  {0.9f,0.8f,0.7f},{0.9f,0.8f,0.7f},{0.8f,0.7f,0.6f},{0.9f,0.8f,0.7f},
  {0.9f,0.8f,0.7f}};
__constant__ float kBruxScore[13] = {2.94f,1.76f,1.76f,1.47f,0.29f,0.88f,0.01f,
                                     0.29f,0.29f,0.01f,0.29f,0.01f,0.01f};

// 64-bit async-copy payload type + address-space-qualified pointer typedefs.
typedef int v2i __attribute__((ext_vector_type(2)));
typedef __attribute__((address_space(1))) v2i*  gptr_v2i;  // global
typedef __attribute__((address_space(3))) v2i*  lptr_v2i;  // LDS
typedef __attribute__((address_space(3))) float* lptr_f;

__device__ __forceinline__ void async_copy_b64(const float* g, float* l) {
#if defined(__AMDGCN__) && __has_builtin(__builtin_amdgcn_global_load_async_to_lds_b64)
  __builtin_amdgcn_global_load_async_to_lds_b64(
      (gptr_v2i)(g), (lptr_v2i)(l), /*imm offset*/0, /*cpol*/0);
#else
#if defined(__AMDGCN__)
#warning "gfx1250 global_load_async_to_lds_b64 builtin unavailable -- plain LDS copy fallback in use"
#endif
  l[0] = g[0];
  l[1] = g[1];
#endif
}

__device__ __forceinline__ void wait_async0() {
#if defined(__AMDGCN__) && __has_builtin(__builtin_amdgcn_s_wait_asynccnt)
  __builtin_amdgcn_s_wait_asynccnt(0);
#elif defined(__AMDGCN__)
  asm volatile("s_wait_asynccnt 0" ::: "memory");
#endif
}

#if HAVE_TDM
typedef unsigned int v4u __attribute__((ext_vector_type(4)));
typedef int          v8i __attribute__((ext_vector_type(8)));
typedef int          v4i __attribute__((ext_vector_type(4)));

// One 1-D TDM descriptor: move n fp32 elements global -> LDS.
// D# layout per CDNA5 ISA ch.8 (groups 0..3); degenerate upper dims set to 1.
// This toolchain's builtin takes 6 args: (g0, g1, g2, g3, extra_v8i, cpol).
__device__ __forceinline__ void tdm_load_1d(const float* gsrc, float* ldst,
                                            uint32_t n /* < 65536 */) {
  const uint64_t ga = (uint64_t)(uintptr_t)gsrc;
  const uint32_t la = (uint32_t)(size_t)(lptr_f)ldst;   // LDS byte offset

  v4u g0;
  g0.x = 1u;                                    // count=1 (valid), user mode
  g0.y = la;                                    // lds_addr (bytes)
  g0.z = (uint32_t)ga;                          // global_addr[31:0]
  g0.w = (uint32_t)((ga >> 32) & 0x01FFFFFFu)   // global_addr[56:32]
       | (2u << 30);                            // type = 2 ("image")

  v8i g1;
  g1[0] = (int)(2u << 16);                      // wg_mask=0 (no cluster), data_size=2 -> 4B
  g1[1] = (int)((n & 0xFFFFu) << 16);           // tensor_dim0[15:0]   (bits 63:48)
  g1[2] = (int)((n >> 16) & 0xFFFFu)            // tensor_dim0[31:16]  (bits 79:64)
        | (int)(1u << 16);                      // tensor_dim1 = 1     (bits 95:80)
  g1[3] = (int)((n & 0xFFFFu) << 16);           // tensor_dim1 hi=0; tile_dim0 = n (bits 127:112)
  g1[4] = 0;                                    // tile_dim1=0 (unused), tile_dim2=0 (unused)
  g1[5] = (int)n;                               // tensor_dim0_stride[31:0]
  g1[6] = 0;                                    // stride0 hi16 = 0, tensor_dim1_stride lo = 0
  g1[7] = 0;                                    // tensor_dim1_stride hi = 0

  v4i g2;
  g2[0] = 1;                                    // tensor_dim2 = 1
  g2[1] = 1;                                    // tensor_dim3 = 1
  g2[2] = 0;                                    // tensor_dim2_stride lo
  g2[3] = 0;                                    // stride hi, tile_dim3 = 0 (unused)

  v4i g3;
  g3[0] = 0;                                    // tensor_dim3_stride lo
  g3[1] = (int)(1u << 16);                      // stride hi = 0, tensor_dim4 = 1 (lo16)
  g3[2] = 0;                                    // tensor_dim4 hi = 0, tile_dim4 = 0
  g3[3] = 0;                                    // reserved

  v8i g4 = {0, 0, 0, 0, 0, 0, 0, 0};            // unused extra group (zero-inert)

  __builtin_amdgcn_tensor_load_to_lds(g0, g1, g2, g3, g4, /*cpol*/0);
}
#endif

__global__ void __launch_bounds__(NTHREADS)
sleep_event_classify_kernel(const float* __restrict__ score,
                            float* __restrict__ out, int B) {
  __shared__ float sm[BATCH_FLOATS];                       // 87528 B frame block
  __shared__ float s_bfj[NFRAMES], s_bfs[NFRAMES];
  __shared__ float s_cfj[NFRAMES], s_cfs[NFRAMES];

  const int b    = blockIdx.x;
  const int tid  = threadIdx.x;
  const int wave = tid >> 5;
  const int lane = tid & 31;
  const float* gsrc = score + (size_t)b * BATCH_FLOATS;

  // ---- Phase 1: dual-engine staging: TDM DMA (chunk A) + async b64 (chunk B)
#if HAVE_TDM
  if (wave == 0)
    tdm_load_1d(gsrc, &sm[0], TDM_FLOATS);     // 1 instruction moves 43768 B
#endif
  for (int i = tid; i < ASYNC_B64; i += NTHREADS)
    async_copy_b64(gsrc + TDM_FLOATS + 2 * i, &sm[TDM_FLOATS + 2 * i]);
  wait_async0();
#if HAVE_TDM
  if (wave == 0)
    __builtin_amdgcn_s_wait_tensorcnt(0);
#endif
  __syncthreads();

  // ---- Phase 2: per-frame ranks + combo logic (wave per frame, strided) ----
  for (int f = wave; f < NFRAMES; f += NWAVES) {
    const float* row = &sm[f * NCLASS];
    const float s0 = row[0], s38 = row[38], s42v = row[42];
    int g0 = 0, g38 = 0, g42 = 0, e38 = 0, e42 = 0;
    for (int c = lane; c < NCLASS; c += 32) {
      const float v = row[c];
      g0  += (v > s0);
      g38 += (v > s38);
      g42 += (v > s42v);
      e38 += ((v == s38) & (c < 38));   // top_k tie-break: lower index wins
      e42 += ((v == s42v) & (c < 42));
    }
#pragma unroll
    for (int m = 16; m >= 1; m >>= 1) {
      g0  += __shfl_xor(g0,  m, 32);
      g38 += __shfl_xor(g38, m, 32);
      g42 += __shfl_xor(g42, m, 32);
      e38 += __shfl_xor(e38, m, 32);
      e42 += __shfl_xor(e42, m, 32);
    }
    if (lane == 0) {
      const bool t0  = g0 < 10;                 // class 0 in top-10
      const bool t38 = (g38 + e38) < 10;        // class 38 in top-10
      const bool t42 = (g42 + e42) < 10;        // class 42 in top-10
      bool any = false; int first = 0;
#pragma unroll
      for (int k = 0; k < 13; ++k) {
        bool p = true;
#pragma unroll
        for (int j = 0; j < 3; ++j) {
          const float v = row[kCls[k][j]];
          p = p && (v >= kLo[k][j]) && (v <= kHi[k][j]);
        }
        if (p && !any) { any = true; first = k; }
      }
      const bool has_nb = t38 || t42 || t0;
      const bool bfj = any && !has_nb;
      s_bfj[f] = bfj ? 1.0f : 0.0f;
      s_bfs[f] = bfj ? kBruxScore[first] : 0.0f;
      const bool cpass = (s42v >= 0.1f) && (s42v <= 1.0f);
      s_cfj[f] = (cpass || t42) ? 1.0f : 0.0f;
      s_cfs[f] = cpass ? 1.5f : (t42 ? 1.0f : 0.0f);
    }
  }
  __syncthreads();

  // ---- Phase 3: group classify (wave 0; lane-parallel windows) ----
  if (wave == 0) {
    const float NEG = -__builtin_inff();
    int   bc = 0, cc = 0;
    float b1 = NEG, b2 = NEG, c1 = NEG, c2 = NEG;
    for (int w = lane; w < 40; w += 32) {
      if (w < 39) {                 // brux: gsize=4, R=39, min_valid=2
        int tc = 0; float ss = 0.0f;
#pragma unroll
        for (int j = 0; j < 4; ++j) {
          const bool jd = s_bfj[w + j] > 0.5f;
          tc += jd; ss += jd ? s_bfs[w + j] : 0.0f;
        }
        const bool gj = tc >= 2;
        bc += gj;
        const float m = gj ? ss : NEG;
        if (m > b1) { b2 = b1; b1 = m; } else if (m > b2) { b2 = m; }
      }
      {                             // cough: gsize=3, R=40, min_valid=2
        int tc = 0; float ss = 0.0f;
#pragma unroll
        for (int j = 0; j < 3; ++j) {
          const bool jd = s_cfj[w + j] > 0.5f;
          tc += jd; ss += jd ? s_cfs[w + j] : 0.0f;
        }
        const bool gj = tc >= 2;
        cc += gj;
        const float m = gj ? ss : NEG;
        if (m > c1) { c2 = c1; c1 = m; } else if (m > c2) { c2 = m; }
      }
    }
    // butterfly all-reduce: counts + (top1, top2) merge
#pragma unroll
    for (int m = 16; m >= 1; m >>= 1) {
      bc += __shfl_xor(bc, m, 32);
      cc += __shfl_xor(cc, m, 32);
      float o1 = __shfl_xor(b1, m, 32), o2 = __shfl_xor(b2, m, 32);
      float n1 = fmaxf(b1, o1);
      float n2 = (b1 >= o1) ? fmaxf(b2, o1) : fmaxf(b1, o2);
      b1 = n1; b2 = n2;
      o1 = __shfl_xor(c1, m, 32); o2 = __shfl_xor(c2, m, 32);
      n1 = fmaxf(c1, o1);
      n2 = (c1 >= o1) ? fmaxf(c2, o1) : fmaxf(c1, o2);
      c1 = n1; c2 = n2;
    }
    if (lane == 0) {
      const bool bok = bc >= 2, cok = cc >= 2;
      out[b]           = bok ? 1.0f : 0.0f;       // brux_j
      out[B + b]       = bok ? (b1 + b2) : 0.0f;  // brux_s
      out[2 * B + b]   = cok ? 1.0f : 0.0f;       // cough_j
      out[3 * B + b]   = cok ? (c1 + c2) : 0.0f;  // cough_s
    }
  }
}

extern "C" void kernel_launch(void* const* d_in, const int* in_sizes, int n_in,
                              void* d_out, int out_size, void* d_ws, size_t ws_size,
                              hipStream_t stream) {
  (void)n_in; (void)out_size; (void)d_ws; (void)ws_size;
  const float* score = (const float*)d_in[0];
  float* out = (float*)d_out;
  const int B = in_sizes[0] / BATCH_FLOATS;   // 2048
  hipLaunchKernelGGL(sleep_event_classify_kernel, dim3(B), dim3(NTHREADS), 0,
                     stream, score, out, B);
}